// LavitTokenCompressor_81965155876927
// MI455X (gfx1250) — compile-verified
//
#include <hip/hip_runtime.h>
#include <hip/hip_bf16.h>
#include <math.h>

// MI455X (gfx1250) implementation.
// Memory-bound problem: >= 1 GB mandatory traffic @ 23.3 TB/s ~= 43us floor.
// GEMM reduced to 32768 x 2048 x 64 fp32 via per-batch factorization of the
// broadcast "global" half; executed with V_WMMA_F32_16X16X4_F32.
// Round 2: weight tile staged TRANSPOSED in LDS so B fragments load as single
// ds_load_b64 into even VGPR pairs (kills the v_dual_mov repacking seen in
// round-1 asm).

typedef __attribute__((ext_vector_type(2))) float v2f;
typedef __attribute__((ext_vector_type(8))) float v8f;

#define BATCH 32
#define NTOK  1024
#define CDIM  4096
#define DDIM  64
#define HALFC 2048
#define LNEPS 1e-5f
#define NTOTAL (BATCH * NTOK)   // 32768 tokens

// ---------------------------------------------------------------- pass 1
// Per-token LayerNorm stats: mu, rstd. One block per token.
__global__ __launch_bounds__(256) void lavit_stats(const float* __restrict__ x,
                                                   float* __restrict__ mu,
                                                   float* __restrict__ rstd) {
  const int token = blockIdx.x;
  const int tid = threadIdx.x;
  const float4* x4 = (const float4*)(x + (size_t)token * CDIM);
  float s = 0.f, sq = 0.f;
  for (int i = tid; i < CDIM / 4; i += 256) {
    float4 v = x4[i];
    s  += v.x + v.y + v.z + v.w;
    sq += v.x * v.x + v.y * v.y + v.z * v.z + v.w * v.w;
  }
  for (int off = 16; off >= 1; off >>= 1) {
    s  += __shfl_xor(s,  off, 32);
    sq += __shfl_xor(sq, off, 32);
  }
  __shared__ float ss[8], ssq[8];
  const int wave = tid >> 5, lane = tid & 31;
  if (lane == 0) { ss[wave] = s; ssq[wave] = sq; }
  __syncthreads();
  if (tid == 0) {
    float S = 0.f, SQ = 0.f;
    for (int w = 0; w < 8; ++w) { S += ss[w]; SQ += ssq[w]; }
    float m   = S * (1.f / CDIM);
    float var = SQ * (1.f / CDIM) - m * m;
    mu[token]   = m;
    rstd[token] = rsqrtf(var + LNEPS);
  }
}

// ---------------------------------------------------------------- pass 2
// gsum[b][c] = sum_n x[b,n,2048+c] * rstd[b,n]
__global__ __launch_bounds__(256) void lavit_colsum(const float* __restrict__ x,
                                                    const float* __restrict__ rstd,
                                                    float* __restrict__ gsum) {
  const int b     = blockIdx.x >> 3;
  const int chunk = blockIdx.x & 7;
  const int c     = chunk * 256 + threadIdx.x;   // 0..2047
  __shared__ float rs[NTOK];
  for (int i = threadIdx.x; i < NTOK; i += 256) rs[i] = rstd[b * NTOK + i];
  __syncthreads();
  const float* xb = x + (size_t)b * NTOK * CDIM + HALFC + c;
  float acc = 0.f;
  for (int n = 0; n < NTOK; ++n) acc += xb[(size_t)n * CDIM] * rs[n];
  gsum[b * HALFC + c] = acc;
}

// ---------------------------------------------------------------- pass 3
// gc[b][d] = global_x[b,:] @ w1[2048:,:] + b1   (64 values per batch)
__global__ __launch_bounds__(64) void lavit_gcontrib(const float* __restrict__ mu,
                                                     const float* __restrict__ rstd,
                                                     const float* __restrict__ gsum,
                                                     const float* __restrict__ lnw,
                                                     const float* __restrict__ lnb,
                                                     const float* __restrict__ w1,
                                                     const float* __restrict__ b1,
                                                     float* __restrict__ gc) {
  const int b = blockIdx.x;
  const int tid = threadIdx.x;   // 64 threads
  __shared__ float gx[HALFC];
  __shared__ float sred[64];
  float sp = 0.f;
  for (int n = tid; n < NTOK; n += 64) sp += mu[b * NTOK + n] * rstd[b * NTOK + n];
  sred[tid] = sp;
  for (int off = 32; off >= 1; off >>= 1) {
    __syncthreads();
    if (tid < off) sred[tid] += sred[tid + off];
  }
  __syncthreads();
  const float S = sred[0];
  for (int c = tid; c < HALFC; c += 64) {
    gx[c] = (gsum[b * HALFC + c] - S) * (1.f / NTOK) * lnw[HALFC + c] + lnb[HALFC + c];
  }
  __syncthreads();
  float acc = b1[tid];
  const float* w1h = w1 + (size_t)HALFC * DDIM + tid;
  for (int c = 0; c < HALFC; ++c) acc += gx[c] * w1h[(size_t)c * DDIM];
  gc[b * DDIM + tid] = acc;
}

// ---------------------------------------------------------------- pass 4
// WMMA GEMM: h = gelu(xn_lo @ w1_top + gc[b]); logits = h@w2+b2;
// keep = (l0+g0 >= l1+g1). 8 waves/block, 16 tokens/wave, K tiled by 64.
__global__ __launch_bounds__(256) void lavit_gemm_keep(
    const float* __restrict__ x, const float* __restrict__ mu,
    const float* __restrict__ rstd, const float* __restrict__ lnw,
    const float* __restrict__ lnb, const float* __restrict__ w1,
    const float* __restrict__ gc, const float* __restrict__ w2,
    const float* __restrict__ b2, const float* __restrict__ gumbel,
    float* __restrict__ keep) {
  // k-stride 68 floats (272B): 16B-aligned rows; fragment reads hit banks
  // (4*row + k) mod 64 -> conflict-free across the 32 lanes.
  __shared__ float sBT[64][68];       // w1 chunk TRANSPOSED: [d][k]
  __shared__ float sA[8][16][68];     // xn tile per wave:    [tok][k]
  const int tid  = threadIdx.x;
  const int wave = tid >> 5;
  const int lane = tid & 31;
  const int wtb  = blockIdx.x * 128 + wave * 16;  // wave token base
  const int b    = (blockIdx.x * 128) >> 10;      // batch (1024 tok/batch)

  // A staging assignment: lane pair per row, 32 contiguous cols per lane.
  const int arow  = lane >> 1;
  const int ahalf = lane & 1;
  const float m_row = mu[wtb + arow];
  const float r_row = rstd[wtb + arow];
  const float* xrow = x + (size_t)(wtb + arow) * CDIM;

  const int nloc  = lane & 15;          // N / fragment-row index
  const int koffb = (lane >> 4) << 1;   // K offset 0 or 2 per lane half

  v8f acc0 = {0,0,0,0,0,0,0,0};
  v8f acc1 = {0,0,0,0,0,0,0,0};
  v8f acc2 = {0,0,0,0,0,0,0,0};
  v8f acc3 = {0,0,0,0,0,0,0,0};

  const float* aRow  = &sA[wave][nloc][0];
  const float* bRow0 = &sBT[nloc][0];
  const float* bRow1 = &sBT[16 + nloc][0];
  const float* bRow2 = &sBT[32 + nloc][0];
  const float* bRow3 = &sBT[48 + nloc][0];

  for (int k0 = 0; k0 < HALFC; k0 += 64) {
    // --- stage B transposed: sBT[d][k] = w1[k0+k][d]
    {
      const int r  = tid >> 2;            // k row 0..63
      const int cq = (tid & 3) << 4;      // d base 0,16,32,48
      const float4* src = (const float4*)(w1 + (size_t)(k0 + r) * DDIM + cq);
      float4 v0 = src[0], v1 = src[1], v2 = src[2], v3 = src[3];
      sBT[cq + 0][r]  = v0.x;  sBT[cq + 1][r]  = v0.y;
      sBT[cq + 2][r]  = v0.z;  sBT[cq + 3][r]  = v0.w;
      sBT[cq + 4][r]  = v1.x;  sBT[cq + 5][r]  = v1.y;
      sBT[cq + 6][r]  = v1.z;  sBT[cq + 7][r]  = v1.w;
      sBT[cq + 8][r]  = v2.x;  sBT[cq + 9][r]  = v2.y;
      sBT[cq + 10][r] = v2.z;  sBT[cq + 11][r] = v2.w;
      sBT[cq + 12][r] = v3.x;  sBT[cq + 13][r] = v3.y;
      sBT[cq + 14][r] = v3.z;  sBT[cq + 15][r] = v3.w;
    }
    // --- stage A: normalized x tile for this wave's 16 tokens
    {
      const int cbase = k0 + ahalf * 32;
      const float4* xs  = (const float4*)(xrow + cbase);
      const float4* w4  = (const float4*)(lnw + cbase);
      const float4* bb4 = (const float4*)(lnb + cbase);
      float* dstA = &sA[wave][arow][ahalf * 32];
      if (k0 + 64 < HALFC) __builtin_prefetch(xrow + cbase + 64, 0, 0);
#pragma unroll
      for (int j = 0; j < 8; ++j) {
        float4 xv = xs[j], wv = w4[j], bv = bb4[j], o;
        o.x = (xv.x - m_row) * r_row * wv.x + bv.x;
        o.y = (xv.y - m_row) * r_row * wv.y + bv.y;
        o.z = (xv.z - m_row) * r_row * wv.z + bv.z;
        o.w = (xv.w - m_row) * r_row * wv.w + bv.w;
        *(float4*)(dstA + j * 4) = o;
      }
    }
    __syncthreads();
    // --- 16 K-steps of 4, 4 N-tiles of 16 (D=64); all operands are single
    // 8B-aligned ds_load_b64 straight into even VGPR pairs.
#pragma unroll
    for (int kk = 0; kk < 64; kk += 4) {
      const int ka = kk + koffb;
      v2f a   = *(const v2f*)(aRow  + ka);
      v2f bf0 = *(const v2f*)(bRow0 + ka);
      v2f bf1 = *(const v2f*)(bRow1 + ka);
      v2f bf2 = *(const v2f*)(bRow2 + ka);
      v2f bf3 = *(const v2f*)(bRow3 + ka);
      acc0 = __builtin_amdgcn_wmma_f32_16x16x4_f32(false, a, false, bf0, (short)0, acc0, false, false);
      acc1 = __builtin_amdgcn_wmma_f32_16x16x4_f32(false, a, false, bf1, (short)0, acc1, false, false);
      acc2 = __builtin_amdgcn_wmma_f32_16x16x4_f32(false, a, false, bf2, (short)0, acc2, false, false);
      acc3 = __builtin_amdgcn_wmma_f32_16x16x4_f32(false, a, false, bf3, (short)0, acc3, false, false);
    }
    __syncthreads();
  }

  // --- epilogue: + gc, exact gelu, 64->2 matmul, keep mask
  const int hi = lane >> 4;
  float gcv[4], w2a[4], w2b[4];
#pragma unroll
  for (int t = 0; t < 4; ++t) {
    const int d = t * 16 + nloc;
    gcv[t] = gc[b * DDIM + d];
    w2a[t] = w2[d * 2 + 0];
    w2b[t] = w2[d * 2 + 1];
  }
  const float bias0 = b2[0], bias1 = b2[1];
#pragma unroll
  for (int v = 0; v < 8; ++v) {
    float h0 = acc0[v] + gcv[0]; h0 = 0.5f * h0 * (1.f + erff(h0 * 0.70710678f));
    float h1 = acc1[v] + gcv[1]; h1 = 0.5f * h1 * (1.f + erff(h1 * 0.70710678f));
    float h2 = acc2[v] + gcv[2]; h2 = 0.5f * h2 * (1.f + erff(h2 * 0.70710678f));
    float h3 = acc3[v] + gcv[3]; h3 = 0.5f * h3 * (1.f + erff(h3 * 0.70710678f));
    float l0 = h0 * w2a[0] + h1 * w2a[1] + h2 * w2a[2] + h3 * w2a[3];
    float l1 = h0 * w2b[0] + h1 * w2b[1] + h2 * w2b[2] + h3 * w2b[3];
    for (int off = 8; off >= 1; off >>= 1) {   // reduce within 16-lane half
      l0 += __shfl_xor(l0, off, 32);
      l1 += __shfl_xor(l1, off, 32);
    }
    if (nloc == 0) {
      const int token = wtb + v + 8 * hi;      // C/D layout: M = v + 8*hi
      const float g0 = gumbel[(size_t)token * 2 + 0];
      const float g1 = gumbel[(size_t)token * 2 + 1];
      keep[token] = ((l0 + bias0 + g0) >= (l1 + bias1 + g1)) ? 1.f : 0.f;
    }
  }
}

// ---------------------------------------------------------------- pass 5
// out = x * keep (float4 grid-stride; 1024 float4 per token)
__global__ __launch_bounds__(256) void lavit_mask(const float* __restrict__ x,
                                                  const float* __restrict__ keep,
                                                  float* __restrict__ out,
                                                  long long nf4) {
  long long i = (long long)blockIdx.x * blockDim.x + threadIdx.x;
  const long long stride = (long long)gridDim.x * blockDim.x;
  const float4* x4 = (const float4*)x;
  float4* o4 = (float4*)out;
  for (; i < nf4; i += stride) {
    const float k = keep[(int)(i >> 10)];
    float4 v = x4[i];
    v.x *= k; v.y *= k; v.z *= k; v.w *= k;
    o4[i] = v;
  }
}

extern "C" void kernel_launch(void* const* d_in, const int* in_sizes, int n_in,
                              void* d_out, int out_size, void* d_ws, size_t ws_size,
                              hipStream_t stream) {
  const float* x      = (const float*)d_in[0];
  const float* gumbel = (const float*)d_in[1];
  const float* lnw    = (const float*)d_in[2];
  const float* lnb    = (const float*)d_in[3];
  const float* w1     = (const float*)d_in[4];
  const float* b1     = (const float*)d_in[5];
  const float* w2     = (const float*)d_in[6];
  const float* b2     = (const float*)d_in[7];
  float* out = (float*)d_out;

  // workspace layout (floats): mu | rstd | gsum | gc | keep  (~640 KB total)
  float* ws   = (float*)d_ws;
  float* mu   = ws;
  float* rstd = mu + NTOTAL;
  float* gsum = rstd + NTOTAL;
  float* gc   = gsum + BATCH * HALFC;
  float* keep = gc + BATCH * DDIM;

  lavit_stats<<<NTOTAL, 256, 0, stream>>>(x, mu, rstd);
  lavit_colsum<<<BATCH * 8, 256, 0, stream>>>(x, rstd, gsum);
  lavit_gcontrib<<<BATCH, 64, 0, stream>>>(mu, rstd, gsum, lnw, lnb, w1, b1, gc);
  lavit_gemm_keep<<<NTOTAL / 128, 256, 0, stream>>>(x, mu, rstd, lnw, lnb, w1,
                                                    gc, w2, b2, gumbel, keep);
  lavit_mask<<<8192, 256, 0, stream>>>(x, keep, out, (long long)NTOTAL * (CDIM / 4));
}